// PlanningModule_34677565948773
// MI455X (gfx1250) — compile-verified
//
#include <hip/hip_runtime.h>

typedef __attribute__((ext_vector_type(2))) float v2f;
typedef __attribute__((ext_vector_type(8))) float v8f;

#define D      2048
#define DH     1024
#define SEQ    4096
#define NB     8

// Fused skinny GEMM: blocks [0,128) -> plan tiles (N over W_prop rows),
// blocks [128,192) -> hid tiles (N over W1 rows).
// Each block: 128 threads = 4 waves; wave w covers K-slice [w*512, w*512+512).
// A rows 8..15 are clamped to rows 0..7 (valid memory, finite values); the
// resulting garbage lives only in C rows 8..15, which are never read.
__global__ __launch_bounds__(128) void planning_gemm_kernel(
    const float* __restrict__ h,
    const float* __restrict__ Wprop,
    const float* __restrict__ bprop,
    const float* __restrict__ W1,
    const float* __restrict__ b1,
    float* __restrict__ out,      // plan: 8 x 2048
    float* __restrict__ hid)      // ws:   8 x 1024
{
    __shared__ float red[4 * 8 * 16];   // per-wave 8x16 partial tiles

    const int tile   = blockIdx.x;
    const int tid    = threadIdx.x;
    const int wave   = tid >> 5;
    const int lane   = tid & 31;
    const int l15    = lane & 15;
    const int hi     = lane >> 4;            // 0: K pair {0,1}; 1: K pair {2,3}
    const bool isPlan = (tile < 128);
    const int n0     = (isPlan ? tile : tile - 128) * 16;
    const float* W   = isPlan ? Wprop : W1;

    // Fold the per-lane K offset into the base pointers so the loop itself
    // has a uniform (scalar) trip count -> scalar backedge, unrollable.
    const int kBase = wave * 512 + hi * 2;
    const float* aPtr = h + ((size_t)(l15 & 7) * SEQ + (SEQ - 1)) * D + kBase;
    const float* bPtr = W + (size_t)(n0 + l15) * D + kBase;

    v8f c0 = {};
    v8f c1 = {};

#pragma unroll 4
    for (int i = 0; i < 512; i += 8) {
        v2f a0 = *(const v2f*)(aPtr + i);
        v2f b0 = *(const v2f*)(bPtr + i);
        v2f a1 = *(const v2f*)(aPtr + i + 4);
        v2f b1 = *(const v2f*)(bPtr + i + 4);
        c0 = __builtin_amdgcn_wmma_f32_16x16x4_f32(
                false, a0, false, b0, (short)0, c0, false, false);
        c1 = __builtin_amdgcn_wmma_f32_16x16x4_f32(
                false, a1, false, b1, (short)0, c1, false, false);
    }

    // C layout: VGPR j, lanes 0..15 hold M=j (N=lane); lanes 16..31 hold M=j+8 (junk).
    if (hi == 0) {
#pragma unroll
        for (int j = 0; j < 8; ++j)
            red[wave * 128 + j * 16 + l15] = c0[j] + c1[j];
    }
    __syncthreads();

    {
        const int mm = tid >> 4;      // 0..7
        const int nn = tid & 15;      // 0..15
        const int i  = mm * 16 + nn;
        float s = red[i] + red[128 + i] + red[256 + i] + red[384 + i];
        const int ng = n0 + nn;
        if (isPlan) {
            const float rootv = h[((size_t)mm * SEQ + (SEQ - 1)) * D + ng];
            out[mm * D + ng] = s + bprop[ng] + rootv;
        } else {
            const float v = s + b1[ng];
            hid[mm * DH + ng] = v > 0.0f ? v : 0.0f;
        }
    }
}

// value[m] = dot(hid[m], W2) + b2 ; 8 waves, one per batch row, wave32 shuffle reduce.
__global__ __launch_bounds__(256) void planning_value_kernel(
    const float* __restrict__ hid,
    const float* __restrict__ W2,
    const float* __restrict__ b2,
    float* __restrict__ out)
{
    const int wave = threadIdx.x >> 5;
    const int lane = threadIdx.x & 31;
    float s = 0.0f;
#pragma unroll 4
    for (int i = lane; i < DH; i += 32)
        s += hid[wave * DH + i] * W2[i];
#pragma unroll
    for (int off = 16; off > 0; off >>= 1)
        s += __shfl_xor(s, off, 32);
    if (lane == 0)
        out[NB * D + wave] = s + b2[0];
}

extern "C" void kernel_launch(void* const* d_in, const int* in_sizes, int n_in,
                              void* d_out, int out_size, void* d_ws, size_t ws_size,
                              hipStream_t stream) {
    const float* h     = (const float*)d_in[0];
    const float* Wprop = (const float*)d_in[1];
    const float* bprop = (const float*)d_in[2];
    const float* W1    = (const float*)d_in[3];
    const float* b1    = (const float*)d_in[4];
    const float* W2    = (const float*)d_in[5];
    const float* b2    = (const float*)d_in[6];

    float* out = (float*)d_out;          // [8*2048 plan | 8 value]
    float* hid = (float*)d_ws;           // 8 x 1024 f32 scratch

    planning_gemm_kernel<<<192, 128, 0, stream>>>(h, Wprop, bprop, W1, b1, out, hid);
    planning_value_kernel<<<1, 256, 0, stream>>>(hid, W2, b2, out);
}